// MemoryModule_541165879332
// MI455X (gfx1250) — compile-verified
//
#include <hip/hip_runtime.h>
#include <hip/hip_bf16.h>

typedef __attribute__((ext_vector_type(2))) float v2f;
typedef __attribute__((ext_vector_type(8))) float v8f;

#define TQ      65536      // tokens (8*8192)
#define CF      256        // feature dim
#define NMEM    256        // memory slots
#define KSPLIT  32
#define KCHUNK  (TQ / KSPLIT)   // 2048
#define LAMBDA  0.0025f
#define EPSV    1e-12f

// ---------------- wave32 reductions ----------------
__device__ __forceinline__ float wave_max(float v) {
#pragma unroll
  for (int o = 16; o; o >>= 1) v = fmaxf(v, __shfl_xor(v, o, 32));
  return v;
}
__device__ __forceinline__ float wave_sum(float v) {
#pragma unroll
  for (int o = 16; o; o >>= 1) v += __shfl_xor(v, o, 32);
  return v;
}

// ---------------- generic Out[r, n] = sum_k A[r,k]*B[n,k], K = 256 ----------------
// block = 256 threads (8 waves); wave w -> 16x16 tile at (blockIdx.x*16, blockIdx.y*128 + 16w)
// f32 WMMA 16x16x4: A lane(l<16): row M=l, K offsets {0,1}; lanes 16..31: K offsets {2,3}.
__global__ void __launch_bounds__(256) gemm_abt_k256(
    const float* __restrict__ A, int lda,
    const float* __restrict__ B, int ldb,
    float* __restrict__ Out, int ldo, int col_off)
{
  const int lane = threadIdx.x & 31;
  const int wave = threadIdx.x >> 5;
  const int r0 = blockIdx.x * 16;
  const int n0 = blockIdx.y * 128 + wave * 16;
  const int rl = lane & 15;
  const int ko = (lane >> 4) << 1;           // 0 for lanes 0-15, 2 for lanes 16-31
  const float* ap = A + (size_t)(r0 + rl) * lda + ko;
  const float* bp = B + (size_t)(n0 + rl) * ldb + ko;
  v8f acc = {0.f, 0.f, 0.f, 0.f, 0.f, 0.f, 0.f, 0.f};
#pragma unroll 16
  for (int k = 0; k < CF; k += 4) {
    v2f a = *(const v2f*)(ap + k);
    v2f b = *(const v2f*)(bp + k);
    acc = __builtin_amdgcn_wmma_f32_16x16x4_f32(false, a, false, b, (short)0, acc,
                                                false, false);
  }
  const int nh = lane >> 4;                  // C/D: VGPR i -> row i + 8*nh, col = lane&15
  const size_t col = (size_t)col_off + n0 + rl;
#pragma unroll
  for (int i = 0; i < 8; ++i)
    Out[(size_t)(r0 + i + 8 * nh) * ldo + col] = acc[i];
}

// ---------------- split-K partials: part[z][r,n] = sum_{k in chunk z} W[r,k]*qT[n,k] ----------------
__global__ void __launch_bounds__(256) gemm_splitk_part(
    const float* __restrict__ Wm,   // [NMEM x TQ] shrunk softmax weights
    const float* __restrict__ qT,   // [CF x TQ]
    float* __restrict__ part)       // [KSPLIT][NMEM*CF]
{
  const int lane = threadIdx.x & 31;
  const int wave = threadIdx.x >> 5;
  const int r0 = blockIdx.x * 16;
  const int n0 = blockIdx.y * 128 + wave * 16;
  const size_t k0 = (size_t)blockIdx.z * KCHUNK;
  const int rl = lane & 15;
  const int ko = (lane >> 4) << 1;
  const float* ap = Wm + (size_t)(r0 + rl) * TQ + k0 + ko;
  const float* bp = qT + (size_t)(n0 + rl) * TQ + k0 + ko;
  v8f acc = {0.f, 0.f, 0.f, 0.f, 0.f, 0.f, 0.f, 0.f};
#pragma unroll 8
  for (int k = 0; k < KCHUNK; k += 4) {
    v2f a = *(const v2f*)(ap + k);
    v2f b = *(const v2f*)(bp + k);
    acc = __builtin_amdgcn_wmma_f32_16x16x4_f32(false, a, false, b, (short)0, acc,
                                                false, false);
  }
  const int nh = lane >> 4;
  float* po = part + (size_t)blockIdx.z * (NMEM * CF);
#pragma unroll
  for (int i = 0; i < 8; ++i)
    po[(size_t)(r0 + i + 8 * nh) * CF + n0 + rl] = acc[i];
}

// deterministic fixed-order reduction of split-K partials, with per-row 1/L1 scale
__global__ void reduce_splitk(const float* __restrict__ part,
                              const float* __restrict__ rowscale,
                              float* __restrict__ am)
{
  const int i = blockIdx.x * 256 + threadIdx.x;   // 0 .. 65535
  float s = 0.f;
#pragma unroll
  for (int z = 0; z < KSPLIT; ++z) s += part[(size_t)z * (NMEM * CF) + i];
  am[i] = s * rowscale[i >> 8];
}

// ---------------- softmax + hard_shrink_relu + L1 stats, rows of length 65536 ----------------
// one block (1024 thr = 32 waves) per row; writes shrunk values in-place, 1/L1 to rowscale
__global__ void __launch_bounds__(1024) rowstats_long(float* __restrict__ S,
                                                      float* __restrict__ rowscale)
{
  __shared__ float red[32];
  const int row = blockIdx.x;
  float* p = S + (size_t)row * TQ;
  const int tid = threadIdx.x;
  const int lane = tid & 31, wv = tid >> 5;

  float m = -3.402823466e38f;
  for (int i = tid; i < TQ; i += 1024) m = fmaxf(m, p[i]);
  m = wave_max(m);
  if (lane == 0) red[wv] = m;
  __syncthreads();
  m = wave_max(red[lane]);
  __syncthreads();

  float z = 0.f;
  for (int i = tid; i < TQ; i += 1024) z += __expf(p[i] - m);
  z = wave_sum(z);
  if (lane == 0) red[wv] = z;
  __syncthreads();
  z = wave_sum(red[lane]);
  __syncthreads();
  const float zinv = 1.f / z;

  float l1 = 0.f;
  for (int i = tid; i < TQ; i += 1024) {
    float pr = __expf(p[i] - m) * zinv;
    float t = pr - LAMBDA;
    float h = fmaxf(t, 0.f) * pr / (fabsf(t) + EPSV);
    p[i] = h;
    l1 += h;
  }
  l1 = wave_sum(l1);
  if (lane == 0) red[wv] = l1;
  __syncthreads();
  l1 = wave_sum(red[lane]);
  if (tid == 0) rowscale[row] = 1.f / fmaxf(l1, EPSV);
}

// ---------------- softmax + shrink + L1-normalize, rows of length 256 ----------------
// one wave per row (8 rows per block); writes normalized weights to Wout
__global__ void __launch_bounds__(256) rowstats_short(const float* __restrict__ S,
                                                      float* __restrict__ Wout)
{
  const int lane = threadIdx.x & 31;
  const int wave = threadIdx.x >> 5;
  const size_t row = (size_t)blockIdx.x * 8 + wave;
  const float* p = S + row * CF;
  float x[8];
#pragma unroll
  for (int j = 0; j < 8; ++j) x[j] = p[lane + 32 * j];
  float m = x[0];
#pragma unroll
  for (int j = 1; j < 8; ++j) m = fmaxf(m, x[j]);
  m = wave_max(m);
  float z = 0.f;
#pragma unroll
  for (int j = 0; j < 8; ++j) { x[j] = __expf(x[j] - m); z += x[j]; }
  z = wave_sum(z);
  const float zinv = 1.f / z;
  float l1 = 0.f;
#pragma unroll
  for (int j = 0; j < 8; ++j) {
    float pr = x[j] * zinv;
    float t = pr - LAMBDA;
    x[j] = fmaxf(t, 0.f) * pr / (fabsf(t) + EPSV);
    l1 += x[j];
  }
  l1 = wave_sum(l1);
  const float sc = 1.f / fmaxf(l1, EPSV);
  float* o = Wout + row * CF;
#pragma unroll
  for (int j = 0; j < 8; ++j) o[lane + 32 * j] = x[j] * sc;
}

// ---------------- gate / new_mem elementwise (+ transposed copy for later GEMM) ----------------
__global__ void __launch_bounds__(256) newmem_kernel(
    const float* __restrict__ mem, const float* __restrict__ am,
    const float* __restrict__ g1, const float* __restrict__ g2,
    const float* __restrict__ Ub, const float* __restrict__ Wb,
    float* __restrict__ newmem, float* __restrict__ memT)
{
  const int m = blockIdx.x, c = threadIdx.x;
  const int i = m * 256 + c;
  const float xg = g1[i] + g2[i] + Ub[c] + Wb[c];
  const float g = 1.f / (1.f + __expf(-xg));
  const float nm = (1.f - g) * mem[i] + g * am[i];
  newmem[i] = nm;
  memT[c * 256 + m] = nm;
}

// ---------------- q [TQ x 256] -> qT [256 x TQ] LDS tile transpose ----------------
__global__ void __launch_bounds__(1024) transpose_q(const float* __restrict__ q,
                                                    float* __restrict__ qT)
{
  __shared__ float tile[32][33];
  const int t0 = blockIdx.x * 32, c0 = blockIdx.y * 32;
  const int tx = threadIdx.x, ty = threadIdx.y;
  tile[ty][tx] = q[(size_t)(t0 + ty) * CF + c0 + tx];
  __syncthreads();
  qT[(size_t)(c0 + ty) * TQ + t0 + tx] = tile[tx][ty];
}

// ---------------- copy q into read_query[:, 0:256] (float4) ----------------
__global__ void __launch_bounds__(256) copy_q_kernel(const float4* __restrict__ q4,
                                                     float4* __restrict__ rq4)
{
  const size_t idx = (size_t)blockIdx.x * 256 + threadIdx.x;  // TQ*64 total
  const size_t t = idx >> 6;
  const int c4 = (int)(idx & 63);
  rq4[t * 128 + c4] = q4[t * 64 + c4];
}

extern "C" void kernel_launch(void* const* d_in, const int* in_sizes, int n_in,
                              void* d_out, int out_size, void* d_ws, size_t ws_size,
                              hipStream_t stream) {
  (void)in_sizes; (void)n_in; (void)out_size; (void)ws_size;
  const float* q   = (const float*)d_in[0];   // [TQ, 256]
  const float* mem = (const float*)d_in[1];   // [256, 256]
  const float* Uw  = (const float*)d_in[2];   // [256, 256]
  const float* Ub  = (const float*)d_in[3];   // [256]
  const float* Ww  = (const float*)d_in[4];   // [256, 256]
  const float* Wb  = (const float*)d_in[5];   // [256]

  float* out    = (float*)d_out;
  float* rq     = out;                                    // [TQ, 512]
  float* attn_r = out + (size_t)TQ * 512;                 // [TQ, 256]
  float* newmem = attn_r + (size_t)TQ * 256;              // [256, 256]

  float* ws   = (float*)d_ws;
  float* Su   = ws;                                       // [256, TQ]   64 MiB
  float* qT   = ws + (size_t)NMEM * TQ;                   // [256, TQ]   64 MiB (reused as Sr)
  float* Sr   = qT;                                       // [TQ, 256]
  float* rsu  = qT + (size_t)CF * TQ;                     // rowscale_u [256] (pad 1024)
  float* am   = rsu + 1024;                               // add_mem [256*256]
  float* g1   = am + NMEM * CF;
  float* g2   = g1 + NMEM * CF;
  float* memT = g2 + NMEM * CF;
  float* part = memT + NMEM * CF;                         // [KSPLIT][256*256] 8 MiB

  // 0) qT = transpose(q)
  transpose_q<<<dim3(TQ / 32, CF / 32), dim3(32, 32), 0, stream>>>(q, qT);
  // 1) S_u = mem @ q^T   [256 x TQ]
  gemm_abt_k256<<<dim3(NMEM / 16, TQ / 128), 256, 0, stream>>>(mem, CF, q, CF, Su, TQ, 0);
  // 2) softmax + shrink in-place, rowscale_u = 1/L1
  rowstats_long<<<NMEM, 1024, 0, stream>>>(Su, rsu);
  // 3) split-K partials of W_u @ q  (via qT), then deterministic reduce with row scale
  gemm_splitk_part<<<dim3(NMEM / 16, CF / 128, KSPLIT), 256, 0, stream>>>(Su, qT, part);
  reduce_splitk<<<(NMEM * CF) / 256, 256, 0, stream>>>(part, rsu, am);
  // 4) gate GEMMs: g1 = mem @ Uw^T, g2 = add_mem @ Ww^T
  gemm_abt_k256<<<dim3(NMEM / 16, CF / 128), 256, 0, stream>>>(mem, CF, Uw, CF, g1, CF, 0);
  gemm_abt_k256<<<dim3(NMEM / 16, CF / 128), 256, 0, stream>>>(am, CF, Ww, CF, g2, CF, 0);
  // 5) new_mem (output) + transposed copy
  newmem_kernel<<<NMEM, 256, 0, stream>>>(mem, am, g1, g2, Ub, Wb, newmem, memT);
  // 6) S_r = q @ new_mem^T   [TQ x 256]   (overwrites qT region)
  gemm_abt_k256<<<dim3(TQ / 16, NMEM / 128), 256, 0, stream>>>(q, CF, newmem, CF, Sr, NMEM, 0);
  // 7) attn_r = normalize(shrink(softmax(S_r)))  -> output region
  rowstats_short<<<TQ / 8, 256, 0, stream>>>(Sr, attn_r);
  // 8) add_memory = attn_r @ new_mem = attn_r @ (memT)^T -> read_query[:, 256:512]
  gemm_abt_k256<<<dim3(TQ / 16, CF / 128), 256, 0, stream>>>(attn_r, NMEM, memT, NMEM, rq, 512, 256);
  // 9) read_query[:, 0:256] = q
  copy_q_kernel<<<(TQ * 64) / 256, 256, 0, stream>>>((const float4*)q, (float4*)rq);
}